// Block_63642825392255
// MI455X (gfx1250) — compile-verified
//
#include <hip/hip_runtime.h>
#include <stdint.h>

// im2col for x:[4,8,512,512] f32 -> out:[4,8,63*63,16,16] f32
// Pure data movement (~32 MiB in, ~124 MiB out) => HBM-bandwidth bound
// (~7 us floor at 23.3 TB/s). Use the CDNA5 async global<->LDS DMA path:
//   global_load_async_to_lds_b128  (stage 16x512 band into LDS, ASYNCcnt)
//   global_store_async_from_lds_b128 (emit overlapping patches, LDS read by DMA)
// Output stores carry TH=NT (cpol=1): write-once data should not displace the
// re-read input bands from WGP$/L2.

#define AS1 __attribute__((address_space(1)))
#define AS3 __attribute__((address_space(3)))

typedef int v4i __attribute__((ext_vector_type(4)));   // matches builtin param type

constexpr int BH = 16, BW = 16, SH = 8, SW = 8;
constexpr int H = 512, W = 512;
constexpr int NBH = (H - BH) / SH + 1;   // 63
constexpr int NBW = (W - BW) / SW + 1;   // 63
constexpr int LDS_ROW = 528;             // 512 + 16-dword pad: patch reads spread over 64 banks

constexpr int TH_NT_STORE = 1;           // ISA store temporal hint: 1 = NT (non-temporal)

__global__ __launch_bounds__(256)
void im2col_async(const float* __restrict__ x, float* __restrict__ out) {
  __shared__ __align__(16) float tile[BH * LDS_ROW];   // ~33 KB of the 320 KB/WGP

  const int tid = threadIdx.x;
  const int wg  = blockIdx.x;
  const int ph  = wg % NBH;       // vertical patch index (band)
  const int bc  = wg / NBH;       // fused batch*channel

  const float* src = x + (size_t)bc * (H * W) + (size_t)(ph * SH) * W;

  // ---- Stage the 16x512 input band into LDS (async DMA, 2048 x b128) ----
#if __has_builtin(__builtin_amdgcn_global_load_async_to_lds_b128)
  #pragma unroll
  for (int k = 0; k < 8; ++k) {
    const int v    = tid + k * 256;   // float4 index, 2048 total
    const int row  = v >> 5;          // 32 float4 per 512-wide row
    const int colv = v & 31;
    __builtin_amdgcn_global_load_async_to_lds_b128(
        (AS1 v4i*)(src + row * W + colv * 4),
        (AS3 v4i*)(&tile[row * LDS_ROW + colv * 4]),
        /*offset=*/0, /*cpol=*/0);
  }
  #if __has_builtin(__builtin_amdgcn_s_wait_asynccnt)
  __builtin_amdgcn_s_wait_asynccnt(0);
  #else
  asm volatile("s_wait_asynccnt 0" ::: "memory");
  #endif
#else
  #pragma unroll
  for (int k = 0; k < 8; ++k) {
    const int v    = tid + k * 256;
    const int row  = v >> 5;
    const int colv = v & 31;
    *(float4*)&tile[row * LDS_ROW + colv * 4] =
        *(const float4*)(src + row * W + colv * 4);
  }
#endif
  __syncthreads();

  // ---- Emit 63 overlapping 16x16 patches straight from LDS ----
  // 63 patches * 64 float4 = 4032 b128 stores per workgroup; the 2x width
  // overlap is served from LDS, writes are contiguous 512B per 32 lanes.
  float* dst = out + ((size_t)bc * (NBH * NBW) + (size_t)ph * NBW) * (BH * BW);
  for (int idx = tid; idx < NBW * (BH * BW / 4); idx += 256) {
    const int pw  = idx >> 6;          // 64 float4 per patch
    const int rem = idx & 63;
    const int i   = rem >> 2;          // row within patch
    const int jv  = rem & 3;           // float4 within row
    const int    lofs = i * LDS_ROW + pw * SW + jv * 4;
    const size_t gofs = (size_t)pw * (BH * BW) + (size_t)(i * BW + jv * 4);
#if __has_builtin(__builtin_amdgcn_global_store_async_from_lds_b128)
    __builtin_amdgcn_global_store_async_from_lds_b128(
        (AS1 v4i*)(dst + gofs),
        (AS3 v4i*)(&tile[lofs]),
        /*offset=*/0, /*cpol=*/TH_NT_STORE);
#else
    *(float4*)(dst + gofs) = *(const float4*)&tile[lofs];
#endif
  }

  // Drain outstanding async stores before the wave (and its LDS) retires.
#if __has_builtin(__builtin_amdgcn_global_store_async_from_lds_b128)
  #if __has_builtin(__builtin_amdgcn_s_wait_asynccnt)
  __builtin_amdgcn_s_wait_asynccnt(0);
  #else
  asm volatile("s_wait_asynccnt 0" ::: "memory");
  #endif
#endif
}

extern "C" void kernel_launch(void* const* d_in, const int* in_sizes, int n_in,
                              void* d_out, int out_size, void* d_ws, size_t ws_size,
                              hipStream_t stream) {
  const float* x  = (const float*)d_in[0];
  float*      out = (float*)d_out;
  const int bc_total = in_sizes[0] / (H * W);          // B*C = 32
  const int nblocks  = bc_total * NBH;                 // 2016 workgroups
  im2col_async<<<dim3(nblocks), dim3(256), 0, stream>>>(x, out);
}